// CrossEntropyLoss_53738630807682
// MI455X (gfx1250) — compile-verified
//
#include <hip/hip_runtime.h>

// CDNA5 / gfx1250. wave32. Uses V_WMMA_F32_16X16X4_F32 (fp32-exact) to batch
// 16 edge dot-products per wave as diag(A_src · B_dst^T).

typedef __attribute__((ext_vector_type(2))) float v2f;
typedef __attribute__((ext_vector_type(8))) float v8f;

#define FEAT_D 128          // feature dim (K) — fixed by the reference
#define KSTEPS (FEAT_D / 4) // 32 WMMA f32 16x16x4 accumulation steps per tile

__global__ void zero_out_kernel(float* out) { out[0] = 0.0f; }

__global__ __launch_bounds__(256) void edge_bce_wmma_kernel(
    const float* __restrict__ h,
    const int*   __restrict__ src,
    const int*   __restrict__ dst,
    int E, float label, float scale,
    float* __restrict__ out)
{
    const int lane = threadIdx.x & 31;
    const int warp = threadIdx.x >> 5;           // 8 waves / block
    const int tile = blockIdx.x * 8 + warp;      // 16 edges per tile
    const int base = tile * 16;

    float wsum = 0.0f;
    if (base < E) {                              // wave-uniform branch; EXEC stays all-1s
        const int idx16 = lane & 15;             // edge slot within tile (M for A, N for B)
        const int hh    = lane >> 4;             // K half-select: lanes 16-31 hold K+2,K+3
        int e  = base + idx16;
        int ec = (e < E) ? e : (E - 1);          // clamp (last tile may be partial when E % 16 != 0)
        const int srow = src[ec];
        const int drow = dst[ec];

        // Per the 32-bit A(16x4) and B(4x16) VGPR layouts, lane L needs the
        // contiguous float2 h[row][4*kb + 2*hh] for BOTH operands.
        const float* pa = h + (size_t)srow * FEAT_D + 2 * hh;
        const float* pb = h + (size_t)drow * FEAT_D + 2 * hh;

        v8f c = {};
        #pragma unroll 8
        for (int kb = 0; kb < KSTEPS; ++kb) {
            v2f a = *(const v2f*)(pa + 4 * kb);  // global_load_b64
            v2f b = *(const v2f*)(pb + 4 * kb);  // global_load_b64
            // D = A(16x4) * B(4x16) + C   -> v_wmma_f32_16x16x4_f32
            c = __builtin_amdgcn_wmma_f32_16x16x4_f32(
                    /*neg_a=*/false, a, /*neg_b=*/false, b,
                    /*c_mod=*/(short)0, c, /*reuse_a=*/false, /*reuse_b=*/false);
        }

        // Diagonal of the 16x16 f32 accumulator:
        //   e in 0..7  -> lane e      , vgpr e
        //   e in 8..15 -> lane e + 16 , vgpr e - 8      (lanes 24..31)
        // In both cases vgpr index == lane & 7.
        const bool lo = (lane < 8);
        const bool hi = (lane >= 24);
        const int  r  = lane & 7;
        float s = c[0];
        s = (r == 1) ? c[1] : s;
        s = (r == 2) ? c[2] : s;
        s = (r == 3) ? c[3] : s;
        s = (r == 4) ? c[4] : s;
        s = (r == 5) ? c[5] : s;
        s = (r == 6) ? c[6] : s;
        s = (r == 7) ? c[7] : s;

        int eidx = lo ? (base + lane) : (base + lane - 16);
        float contrib = 0.0f;
        if ((lo || hi) && (eidx < E)) {
            // numerically stable BCE with logits
            contrib = fmaxf(s, 0.0f) - s * label + log1pf(expf(-fabsf(s)));
        }
        // wave32 reduction
        #pragma unroll
        for (int off = 16; off > 0; off >>= 1)
            contrib += __shfl_xor(contrib, off, 32);
        wsum = contrib;
    }

    __shared__ float smem[8];
    if (lane == 0) smem[warp] = wsum;
    __syncthreads();
    if (threadIdx.x == 0) {
        float t = 0.0f;
        #pragma unroll
        for (int i = 0; i < 8; ++i) t += smem[i];
        atomicAdd(out, t * scale);               // global_atomic_add_f32, ~1 per block
    }
}

extern "C" void kernel_launch(void* const* d_in, const int* in_sizes, int n_in,
                              void* d_out, int out_size, void* d_ws, size_t ws_size,
                              hipStream_t stream) {
    const float* h       = (const float*)d_in[0];
    const int*   pos_src = (const int*)d_in[1];
    const int*   pos_dst = (const int*)d_in[2];
    const int*   neg_src = (const int*)d_in[3];
    const int*   neg_dst = (const int*)d_in[4];
    float*       out     = (float*)d_out;

    const int Ep = in_sizes[1];
    const int En = in_sizes[3];
    const float scale = 1.0f / (float)((long long)Ep + (long long)En);

    zero_out_kernel<<<1, 1, 0, stream>>>(out);

    const int tiles_p  = (Ep + 15) / 16;
    const int blocks_p = (tiles_p + 7) / 8;
    edge_bce_wmma_kernel<<<blocks_p, 256, 0, stream>>>(h, pos_src, pos_dst, Ep, 1.0f, scale, out);

    const int tiles_n  = (En + 15) / 16;
    const int blocks_n = (tiles_n + 7) / 8;
    edge_bce_wmma_kernel<<<blocks_n, 256, 0, stream>>>(h, neg_src, neg_dst, En, 0.0f, scale, out);
}